// GCNHAG_44461501448670
// MI455X (gfx1250) — compile-verified
//
#include <hip/hip_runtime.h>
#include <math.h>

typedef __attribute__((ext_vector_type(2))) float v2f;
typedef __attribute__((ext_vector_type(8))) float v8f;

// ---------------------------------------------------------------------------
// FP32 WMMA GEMM: C[M x NPAD] = A[M x K] @ B[K x NPAD], row-major.
// Compile-time K and NPAD (NPAD % 16 == 0, B zero-padded) -> branch-free,
// fully unrolled inner loop of V_WMMA_F32_16X16X4_F32.
// Block = NT waves; all waves share one 16xK A tile staged in LDS
// (pitch K+4 to dodge bank conflicts); wave w computes N-tile w.
//   A frag (16x4 f32): lanes 0-15: M=lane, K={k,k+1}; lanes 16-31: K={k+2,k+3}
//   B frag (4x16 f32): symmetric, N striped across lanes.
//   C/D: VGPR i -> row = i + 8*(lane>=16), col = lane&15.
// Requires M % 16 == 0 (M = 100000 here).
// ---------------------------------------------------------------------------
template <int K, int NPAD, int NT>
__global__ __launch_bounds__(32 * NT) void gcn_gemm_wmma_f32(
    const float* __restrict__ A, const float* __restrict__ B,
    float* __restrict__ C) {
    constexpr int KP = K + 4;                 // LDS pitch (floats)
    __shared__ float As[16 * KP];

    const int tid  = threadIdx.x;
    const int m0   = blockIdx.x * 16;

    // Cooperative, coalesced stage of the 16xK A tile into LDS (float4).
    for (int i = tid; i < 16 * (K / 4); i += 32 * NT) {
        const int r  = i / (K / 4);
        const int c4 = (i - r * (K / 4)) * 4;
        const float4 v = *(const float4*)(A + (size_t)(m0 + r) * K + c4);
        *(float4*)(As + r * KP + c4) = v;
    }
    __syncthreads();

    const int wave = tid >> 5;
    const int lane = tid & 31;
    const int half = lane >> 4;               // which K pair of the 4-step
    const int l16  = lane & 15;
    const int n0   = wave * 16;
    const int bcol = n0 + l16;

    v8f acc = {};
#pragma unroll
    for (int k = 0; k < K; k += 4) {
        const int kk = k + 2 * half;
        v2f a, b;
        a.x = As[l16 * KP + kk];
        a.y = As[l16 * KP + kk + 1];
        b.x = B[kk * NPAD + bcol];
        b.y = B[(kk + 1) * NPAD + bcol];
        acc = __builtin_amdgcn_wmma_f32_16x16x4_f32(
            false, a, false, b, (short)0, acc, false, false);
    }

#pragma unroll
    for (int i = 0; i < 8; ++i)
        C[(size_t)(m0 + i + 8 * half) * NPAD + n0 + l16] = acc[i];
}

// ---------------------------------------------------------------------------
// Zero-pad W2 [64x40] -> W2p [64x48]
// ---------------------------------------------------------------------------
__global__ void gcn_pad_w2(const float* __restrict__ W2, float* __restrict__ W2p) {
    int i = blockIdx.x * blockDim.x + threadIdx.x;
    if (i >= 64 * 48) return;
    int k = i / 48, n = i - k * 48;
    W2p[i] = (n < 40) ? W2[k * 40 + n] : 0.0f;
}

// ---------------------------------------------------------------------------
// Degree / normalization
// ---------------------------------------------------------------------------
__global__ void gcn_init_deg(float* __restrict__ deg, int n) {
    int i = blockIdx.x * blockDim.x + threadIdx.x;
    if (i < n) deg[i] = 1.0f;   // self-loop contribution
}

__global__ void gcn_degree(const int* __restrict__ col, float* __restrict__ deg, int nE) {
    int e = blockIdx.x * blockDim.x + threadIdx.x;
    if (e < nE) atomicAdd(deg + col[e], 1.0f);
}

__global__ void gcn_dinv(float* __restrict__ deg, int n) {
    int i = blockIdx.x * blockDim.x + threadIdx.x;
    if (i < n) {
        float d = deg[i];
        deg[i] = (d > 0.0f) ? rsqrtf(d) : 0.0f;   // in place: deg -> dinv
    }
}

// ---------------------------------------------------------------------------
// Output init: out[i,c] = b[c] + h[i,c] * dinv[i]^2   (bias + self-loop edge)
// ---------------------------------------------------------------------------
__global__ void gcn_init_out64(const float* __restrict__ h, const float* __restrict__ dinv,
                               const float* __restrict__ bias, float* __restrict__ out, int n) {
    int gid = blockIdx.x * blockDim.x + threadIdx.x;
    if (gid >= n * 64) return;
    int node = gid >> 6, ch = gid & 63;
    float di = dinv[node];
    out[gid] = bias[ch] + h[gid] * di * di;
}

// h is padded to pitch 48; out is dense [n x 40]
__global__ void gcn_init_out40(const float* __restrict__ h, const float* __restrict__ dinv,
                               const float* __restrict__ bias, float* __restrict__ out, int n) {
    int gid = blockIdx.x * blockDim.x + threadIdx.x;
    if (gid >= n * 40) return;
    int node = gid / 40, ch = gid - node * 40;
    float di = dinv[node];
    out[gid] = bias[ch] + h[(size_t)node * 48 + ch] * di * di;
}

// ---------------------------------------------------------------------------
// Edge scatter: one wave per edge. Gathers h[row], scales by
// dinv[row]*dinv[col], atomically accumulates into out[col]. All the hot
// buffers (h1/out1/h2/out: ~83 MB total) fit in the 192 MB L2, so these
// atomics and gathers never touch HBM.
// ---------------------------------------------------------------------------
__global__ void gcn_scatter64(const int* __restrict__ row, const int* __restrict__ col,
                              const float* __restrict__ dinv, const float* __restrict__ h,
                              float* __restrict__ out, int nE) {
    int gid = blockIdx.x * blockDim.x + threadIdx.x;
    int e = gid >> 5, lane = gid & 31;
    if (e >= nE) return;
    int r = row[e], c = col[e];
    float nrm = dinv[r] * dinv[c];
    const float2* hp = (const float2*)(h + (size_t)r * 64);
    float2 v = hp[lane];
    float* op = out + (size_t)c * 64 + 2 * lane;
    atomicAdd(op + 0, v.x * nrm);
    atomicAdd(op + 1, v.y * nrm);
}

// h has pitch 48 (cols 40..47 are garbage-free zeros from padded GEMM);
// out is dense [n x 40]
__global__ void gcn_scatter40(const int* __restrict__ row, const int* __restrict__ col,
                              const float* __restrict__ dinv, const float* __restrict__ h,
                              float* __restrict__ out, int nE) {
    int gid = blockIdx.x * blockDim.x + threadIdx.x;
    int e = gid >> 5, lane = gid & 31;
    if (e >= nE) return;
    int r = row[e], c = col[e];
    float nrm = dinv[r] * dinv[c];
    const float* hp = h + (size_t)r * 48;
    float* op = out + (size_t)c * 40;
    atomicAdd(op + lane, hp[lane] * nrm);
    if (lane < 8) atomicAdd(op + 32 + lane, hp[32 + lane] * nrm);
}

// ---------------------------------------------------------------------------
// Row-wise log_softmax over 40 channels, one wave per node, in place.
// ---------------------------------------------------------------------------
__global__ void gcn_logsoftmax40(float* __restrict__ out, int n) {
    int gid = blockIdx.x * blockDim.x + threadIdx.x;
    int node = gid >> 5, lane = gid & 31;
    if (node >= n) return;
    float* p = out + (size_t)node * 40;
    float v0 = p[lane];
    float v1 = (lane < 8) ? p[32 + lane] : -INFINITY;
    float m = fmaxf(v0, v1);
#pragma unroll
    for (int off = 16; off > 0; off >>= 1) m = fmaxf(m, __shfl_xor(m, off, 32));
    float s = __expf(v0 - m) + ((lane < 8) ? __expf(v1 - m) : 0.0f);
#pragma unroll
    for (int off = 16; off > 0; off >>= 1) s += __shfl_xor(s, off, 32);
    float ls = __logf(s);
    p[lane] = v0 - m - ls;
    if (lane < 8) p[32 + lane] = v1 - m - ls;
}

// ---------------------------------------------------------------------------
extern "C" void kernel_launch(void* const* d_in, const int* in_sizes, int n_in,
                              void* d_out, int out_size, void* d_ws, size_t ws_size,
                              hipStream_t stream) {
    const float* x  = (const float*)d_in[0];   // [N,128]
    const int*   ei = (const int*)d_in[1];     // [2,E] (JAX x64 off -> int32)
    const float* W1 = (const float*)d_in[3];   // [128,64]
    const float* b1 = (const float*)d_in[4];   // [64]
    const float* W2 = (const float*)d_in[5];   // [64,40]
    const float* b2 = (const float*)d_in[6];   // [40]

    const int N = in_sizes[0] / 128;           // 100000
    const int E = in_sizes[1] / 2;             // 3200000
    const int* row = ei;                       // sources
    const int* col = ei + E;                   // targets

    // Workspace carve-up (256B aligned): dinv | h1 | out1 | h2p | W2p (~71 MB)
    char* ws = (char*)d_ws;
    size_t off = 0;
    auto carve = [&](size_t bytes) -> void* {
        void* p = ws + off;
        off = (off + bytes + 255) & ~(size_t)255;
        return p;
    };
    float* dinv = (float*)carve((size_t)N * 4);
    float* h1   = (float*)carve((size_t)N * 64 * 4);
    float* out1 = (float*)carve((size_t)N * 64 * 4);
    float* h2p  = (float*)carve((size_t)N * 48 * 4);
    float* W2p  = (float*)carve((size_t)64 * 48 * 4);
    float* out2 = (float*)d_out;               // [N,40]

    const int TB = 256;

    // 0) pad W2 to [64x48]
    gcn_pad_w2<<<(64 * 48 + TB - 1) / TB, TB, 0, stream>>>(W2, W2p);

    // 1) degrees (self-loop folded into init) -> dinv, in place
    gcn_init_deg<<<(N + TB - 1) / TB, TB, 0, stream>>>(dinv, N);
    gcn_degree  <<<(E + TB - 1) / TB, TB, 0, stream>>>(col, dinv, E);
    gcn_dinv    <<<(N + TB - 1) / TB, TB, 0, stream>>>(dinv, N);

    // 2) layer 1: h1 = x @ W1 (WMMA f32, K=128, N=64, 4 waves/block)
    gcn_gemm_wmma_f32<128, 64, 4><<<N / 16, 128, 0, stream>>>(x, W1, h1);
    gcn_init_out64<<<((size_t)N * 64 + TB - 1) / TB, TB, 0, stream>>>(h1, dinv, b1, out1, N);
    gcn_scatter64<<<((size_t)E * 32 + TB - 1) / TB, TB, 0, stream>>>(row, col, dinv, h1, out1, E);

    // 3) layer 2: h2p = out1 @ W2p (WMMA f32, K=64, N=48 padded, 3 waves/block)
    gcn_gemm_wmma_f32<64, 48, 3><<<N / 16, 96, 0, stream>>>(out1, W2p, h2p);
    gcn_init_out40<<<((size_t)N * 40 + TB - 1) / TB, TB, 0, stream>>>(h2p, dinv, b2, out2, N);
    gcn_scatter40<<<((size_t)E * 32 + TB - 1) / TB, TB, 0, stream>>>(row, col, dinv, h2p, out2, E);

    // 4) log_softmax over 40 channels, in place on d_out
    gcn_logsoftmax40<<<((size_t)N * 32 + TB - 1) / TB, TB, 0, stream>>>(out2, N);
}